// ChamferLoss_53437983097030
// MI455X (gfx1250) — compile-verified
//
#include <hip/hip_runtime.h>

typedef __attribute__((ext_vector_type(2))) float v2f;
typedef __attribute__((ext_vector_type(8))) float v8f;
typedef int v4i __attribute__((vector_size(4 * sizeof(int))));

#define CH_B 32
#define CH_N 2048
#define CH_M 2048
#define CH_D 128
#define TILE 16
#define WAVES 8
#define ROWS_PER_WG (WAVES * TILE)   // 128 x-rows per workgroup
#define LDSW 132                     // padded LDS row stride (floats) -> no bank conflicts
#define EPS_F 1e-12f
#define DEPS2 (128.0f * EPS_F * EPS_F)
#define INF_BITS 0x7F800000u

#define AS1 __attribute__((address_space(1)))
#define AS3 __attribute__((address_space(3)))

#if __has_builtin(__builtin_amdgcn_global_load_async_to_lds_b128)
#  define USE_ASYNC_LDS 1
#  if __has_builtin(__builtin_amdgcn_s_wait_asynccnt)
#    define WAIT_ASYNC() __builtin_amdgcn_s_wait_asynccnt(0)
#  else
#    define WAIT_ASYNC() asm volatile("s_wait_asynccnt 0" ::: "memory")
#  endif
#else
#  define USE_ASYNC_LDS 0
#endif

// ---------------------------------------------------------------------------
// Pass 0: per-row terms  r = ||p||^2 + epsCoef * sum(p), one wave per row.
// Also initializes the corresponding min-buffer entry to +inf bits.
// ---------------------------------------------------------------------------
__global__ __launch_bounds__(256) void chamfer_rowterm(
    const float* __restrict__ pts, float* __restrict__ rterm,
    unsigned int* __restrict__ minbuf, float epsCoef, int totalRows) {
  const int wrow = (blockIdx.x * blockDim.x + threadIdx.x) >> 5;
  const int lane = threadIdx.x & 31;
  if (wrow >= totalRows) return;
  const float4* p = (const float4*)(pts + (size_t)wrow * CH_D);
  float4 v = p[lane];                       // 32 lanes x 4 floats = 128
  float s2 = v.x * v.x + v.y * v.y + v.z * v.z + v.w * v.w;
  float s1 = v.x + v.y + v.z + v.w;
  #pragma unroll
  for (int off = 16; off >= 1; off >>= 1) {
    s2 += __shfl_xor(s2, off, 32);
    s1 += __shfl_xor(s1, off, 32);
  }
  if (lane == 0) {
    rterm[wrow]  = s2 + epsCoef * s1;
    minbuf[wrow] = INF_BITS;
  }
}

// ---------------------------------------------------------------------------
// Pass 1: fused distance GEMM (V_WMMA_F32_16X16X4_F32) + row/col minima.
//   sq[i,j] = rx[i] + ry[j] + D*eps^2 - 2 * dot(x_i, y_j);  dist = sqrt(max(sq,0))
//   minRow[b,i] = min_j dist (register-accumulated, direct store)
//   minCol[b,j] = min_i dist (LDS-combined, global atomicMin on float bits)
// Y tiles double-buffered in LDS; prefetch via global_load_async_to_lds_b128
// (ASYNCcnt) when available, else a software-pipelined load->compute->store.
// ---------------------------------------------------------------------------
__global__ __launch_bounds__(256) void chamfer_wmma_pass(
    const float* __restrict__ X, const float* __restrict__ Y,
    const float* __restrict__ rx, const float* __restrict__ ry,
    unsigned int* __restrict__ minRow, unsigned int* __restrict__ minCol) {
  __shared__ float ldsY[2][TILE][LDSW];     // double-buffered y tile [16][128+pad]
  __shared__ float ldsRX[WAVES][TILE];
  __shared__ float ldsRY[CH_M];
  __shared__ unsigned int ldsCM[TILE];

  const int tid   = threadIdx.x;
  const int lane  = tid & 31;
  const int wave  = tid >> 5;
  const int nIBlk = CH_N / ROWS_PER_WG;
  const int b     = blockIdx.x / nIBlk;
  const int iblk  = blockIdx.x - b * nIBlk;

  const float* Xb = X + (size_t)b * CH_N * CH_D;
  const float* Yb = Y + (size_t)b * CH_M * CH_D;
  const int i0 = iblk * ROWS_PER_WG + wave * TILE;   // this wave's 16 x-rows

  // stage per-column terms + per-row terms + col-min scratch
  for (int t = tid; t < CH_M; t += 256) ldsRY[t] = ry[(size_t)b * CH_M + t];
  if (lane < 16) ldsRX[wave][lane] = rx[(size_t)b * CH_N + i0 + lane];
  if (tid < TILE) ldsCM[tid] = INF_BITS;

  // A fragments, ISA 16x4 f32 layout: lanes 0-15 -> K=0,1 ; lanes 16-31 -> K=2,3
  const int arow = lane & 15;
  const int ksel = (lane >> 4) << 1;
  v2f af[CH_D / 4];
  {
    const float* ap = Xb + (size_t)(i0 + arow) * CH_D + ksel;
    #pragma unroll
    for (int k = 0; k < CH_D / 4; ++k) { af[k][0] = ap[4 * k]; af[k][1] = ap[4 * k + 1]; }
  }

  // tile-staging coordinates: 256 threads x 8 floats (two b128) = 16x128
  const int sr = tid >> 4;
  const int sc = (tid & 15) * 8;

  // preload first y tile (synchronous; outside the hot loop)
  {
    const float4* s = (const float4*)(Yb + (size_t)sr * CH_D + sc);
    float4 v0 = s[0], v1 = s[1];
    float* d = &ldsY[0][sr][sc];
    d[0] = v0.x; d[1] = v0.y; d[2] = v0.z; d[3] = v0.w;
    d[4] = v1.x; d[5] = v1.y; d[6] = v1.z; d[7] = v1.w;
  }
  __syncthreads();

  float rowAdd[8];
  const int rbase = (lane >> 4) << 3;   // C layout: VGPR v <-> row v (lo) / v+8 (hi)
  #pragma unroll
  for (int v = 0; v < 8; ++v) rowAdd[v] = ldsRX[wave][rbase + v] + DEPS2;

  float rmin[8];
  #pragma unroll
  for (int v = 0; v < 8; ++v) rmin[v] = __builtin_inff();

  const int nJ = CH_M / TILE;
  for (int j = 0; j < nJ; ++j) {
    const int buf = j & 1;

    // ---- issue prefetch of tile j+1 into the other buffer (lands later) ----
#if USE_ASYNC_LDS
    if (j + 1 < nJ) {
      const float* s = Yb + (size_t)((j + 1) * TILE + sr) * CH_D + sc;
      float* d = &ldsY[buf ^ 1][sr][sc];
      __builtin_amdgcn_global_load_async_to_lds_b128(
          (AS1 v4i*)(void*)s, (AS3 v4i*)(void*)d, 0, 0);
      __builtin_amdgcn_global_load_async_to_lds_b128(
          (AS1 v4i*)(void*)(s + 4), (AS3 v4i*)(void*)(d + 4), 0, 0);
    }
#else
    float4 p0, p1;
    if (j + 1 < nJ) {
      const float4* s = (const float4*)(Yb + (size_t)((j + 1) * TILE + sr) * CH_D + sc);
      p0 = s[0];   // loads issue here; waited on only after the WMMA block
      p1 = s[1];
    }
#endif

    // ---- 16x16x128 tile: two independent WMMA accumulator chains ----
    const float* bp = &ldsY[buf][lane & 15][ksel];
    v8f acc0 = {0.f, 0.f, 0.f, 0.f, 0.f, 0.f, 0.f, 0.f};
    v8f acc1 = {0.f, 0.f, 0.f, 0.f, 0.f, 0.f, 0.f, 0.f};
    #pragma unroll
    for (int k = 0; k < CH_D / 8; ++k) {
      v2f bf0, bf1;
      bf0[0] = bp[8 * k];     bf0[1] = bp[8 * k + 1];
      bf1[0] = bp[8 * k + 4]; bf1[1] = bp[8 * k + 5];
      acc0 = __builtin_amdgcn_wmma_f32_16x16x4_f32(false, af[2 * k],     false, bf0,
                                                   (short)0, acc0, false, false);
      acc1 = __builtin_amdgcn_wmma_f32_16x16x4_f32(false, af[2 * k + 1], false, bf1,
                                                   (short)0, acc1, false, false);
    }

    // ---- epilogue: dist + fused min reductions ----
    const float colAdd = ldsRY[j * TILE + (lane & 15)];
    float cmin = __builtin_inff();
    #pragma unroll
    for (int v = 0; v < 8; ++v) {
      float sq = rowAdd[v] + colAdd - 2.0f * (acc0[v] + acc1[v]);
      sq = sq > 0.f ? sq : 0.f;
      float dv = __builtin_sqrtf(sq);
      rmin[v] = fminf(rmin[v], dv);
      cmin = fminf(cmin, dv);
    }
    cmin = fminf(cmin, __shfl_xor(cmin, 16, 32));       // combine row halves
    if (lane < 16) atomicMin(&ldsCM[lane], __float_as_uint(cmin));
    __syncthreads();

#if !USE_ASYNC_LDS
    if (j + 1 < nJ) {   // land the prefetch after compute (hides load latency)
      float* d = &ldsY[buf ^ 1][sr][sc];
      d[0] = p0.x; d[1] = p0.y; d[2] = p0.z; d[3] = p0.w;
      d[4] = p1.x; d[5] = p1.y; d[6] = p1.z; d[7] = p1.w;
    }
#endif
    if (tid < TILE) {
      atomicMin(&minCol[(size_t)b * CH_M + j * TILE + tid], ldsCM[tid]);
      ldsCM[tid] = INF_BITS;
    }
#if USE_ASYNC_LDS
    WAIT_ASYNC();       // async LDS writes complete before publishing barrier
#endif
    __syncthreads();    // tile (buf^1) now visible; buf free for next prefetch
  }

  // row minima: min across the 16 lanes of each half, one writer per row
  #pragma unroll
  for (int v = 0; v < 8; ++v) {
    float r = rmin[v];
    r = fminf(r, __shfl_xor(r, 1, 32));
    r = fminf(r, __shfl_xor(r, 2, 32));
    r = fminf(r, __shfl_xor(r, 4, 32));
    r = fminf(r, __shfl_xor(r, 8, 32));
    if ((lane & 15) == 0)
      minRow[(size_t)b * CH_N + i0 + rbase + v] = __float_as_uint(r);
  }
}

// ---------------------------------------------------------------------------
// Pass 2: single-block final sum of both min buffers into d_out[0].
// ---------------------------------------------------------------------------
__global__ __launch_bounds__(256) void chamfer_reduce(
    const unsigned int* __restrict__ mA, const unsigned int* __restrict__ mB,
    float* __restrict__ out, int n) {
  float s = 0.f;
  for (int i = threadIdx.x; i < n; i += 256)
    s += __uint_as_float(mA[i]) + __uint_as_float(mB[i]);
  #pragma unroll
  for (int off = 16; off >= 1; off >>= 1) s += __shfl_xor(s, off, 32);
  __shared__ float ws[WAVES];
  const int lane = threadIdx.x & 31, wave = threadIdx.x >> 5;
  if (lane == 0) ws[wave] = s;
  __syncthreads();
  if (threadIdx.x == 0) {
    float t = 0.f;
    #pragma unroll
    for (int w = 0; w < WAVES; ++w) t += ws[w];
    out[0] = t;
  }
}

extern "C" void kernel_launch(void* const* d_in, const int* in_sizes, int n_in,
                              void* d_out, int out_size, void* d_ws, size_t ws_size,
                              hipStream_t stream) {
  const float* x = (const float*)d_in[0];   // [B, N, D] fp32
  const float* y = (const float*)d_in[1];   // [B, M, D] fp32

  // workspace layout: rx[B*N] | ry[B*M] | minxy[B*N] | minyx[B*M]  (1 MiB total)
  float* rx = (float*)d_ws;
  float* ry = rx + (size_t)CH_B * CH_N;
  unsigned int* minxy = (unsigned int*)(ry + (size_t)CH_B * CH_M);
  unsigned int* minyx = minxy + (size_t)CH_B * CH_N;

  const int rowsX = CH_B * CH_N;
  const int rowsY = CH_B * CH_M;
  chamfer_rowterm<<<rowsX / 8, 256, 0, stream>>>(x, rx, minxy, 2.0f * EPS_F, rowsX);
  chamfer_rowterm<<<rowsY / 8, 256, 0, stream>>>(y, ry, minyx, -2.0f * EPS_F, rowsY);

  chamfer_wmma_pass<<<CH_B * (CH_N / ROWS_PER_WG), 256, 0, stream>>>(
      x, y, rx, ry, minxy, minyx);

  chamfer_reduce<<<1, 256, 0, stream>>>(minxy, minyx, (float*)d_out, CH_B * CH_N);
}